// ODE_6073083756761
// MI455X (gfx1250) — compile-verified
//
#include <hip/hip_runtime.h>

// ---------------------------------------------------------------------------
// Per-sample-weights MLP (64 -> 512 -> 512 -> 512 -> 64, SiLU), B=512 samples.
//
// Roofline: 1.21 GB of fp32 weights streamed once @ 23.3 TB/s => ~52 us floor,
// needing only ~11.6 TFLOP/s fp32. Pure bandwidth-bound => keep fp32, optimize
// data movement, and run the MACs on V_WMMA_F32_16X16X4_F32 (full-precision
// fp32 WMMA) so VALU is free for SiLU/addressing.
//
// GEMV-as-WMMA mapping (per 16-output tile, K accumulated in steps of 4):
//   D[m,n] = sum_k A[m,k] * B[k,n]
//   A[m,k] = x[k0+k]   (broadcast over all 16 rows m)
//   B[k,n] = W[n,k0+k] (16 output rows along N)
//   => D[m,n] = y[n] for every m; lane n (n<16) of D VGPR0 holds output n.
//
// Codegen verified (round 1): inner loop = s_clause of 4x global_load_b64
// (offsets 0/16/32/48 => lane pair covers 64B contiguous per weight row per
// clause) + ds_load_2addr_b64 + 4x v_wmma_f32_16x16x4_f32 with partial
// loadcnt waits. Memory-bound target met at the instruction level.
// ---------------------------------------------------------------------------

typedef float v2f __attribute__((ext_vector_type(2)));
typedef float v8f __attribute__((ext_vector_type(8)));

#define D_DIM 64
#define H_DIM 512
#define NB    512        // batch = number of workgroups

// per-sample parameter offsets (in floats), PyTorch [out,in] order
#define OFF_W1 0
#define OFF_B1 32768
#define OFF_W2 33280
#define OFF_B2 295424
#define OFF_W3 295936
#define OFF_B3 558080
#define OFF_W4 558592
#define OFF_B4 591360
#define P_DIM  591424

__device__ __forceinline__ float silu_f32(float x) {
    // x * sigmoid(x) with hardware v_rcp_f32 (avoids the correctly-rounded
    // division expansion; epilogue-only, precision ~1 ulp of rcp).
    return x * __builtin_amdgcn_rcpf(1.0f + __expf(-x));
}

// One dense layer: hout[t*16+n] = act( sum_k W[t*16+n,k]*hin[k] + bias[t*16+n] )
// W, bias: global (per-sample). hin: LDS. hout: LDS (or global for last layer).
__device__ __forceinline__ void layer_wmma(
    const float* __restrict__ W,
    const float* __restrict__ bias,
    const float* __restrict__ hin,
    float* __restrict__ hout,
    int K, int nOut, bool do_silu,
    int wave, int nWaves, int lane)
{
    const int half  = lane >> 4;
    const int n     = lane & 15;
    const int koff  = 2 * half;          // 0 or 2
    const int nTile = nOut >> 4;

    for (int tile = wave; tile < nTile; tile += nWaves) {
        const float* wrow = W + (size_t)(tile * 16 + n) * K + koff;
        const float* xv   = hin + koff;
        v8f c = {0.f, 0.f, 0.f, 0.f, 0.f, 0.f, 0.f, 0.f};

        #pragma unroll 4
        for (int k0 = 0; k0 < K; k0 += 4) {
            v2f a = *(const v2f*)(xv + k0);     // activation broadcast (LDS)
            v2f b = *(const v2f*)(wrow + k0);   // W^T fragment (global, 8B/lane)
            // 8 args: (neg_a, A, neg_b, B, c_mod, C, reuse_a, reuse_b)
            c = __builtin_amdgcn_wmma_f32_16x16x4_f32(
                    false, a, false, b, (short)0, c, false, false);
        }

        if (lane < 16) {                        // D VGPR0, lanes 0-15: y[n]
            float y = c[0] + bias[tile * 16 + n];
            hout[tile * 16 + n] = do_silu ? silu_f32(y) : y;
        }
    }
}

__global__ void __launch_bounds__(256, 1)
ode_persample_mlp_kernel(const float* __restrict__ z,
                         const float* __restrict__ params,
                         float* __restrict__ out)
{
    __shared__ float hA[H_DIM];   // activation ping
    __shared__ float hB[H_DIM];   // activation pong

    const int sample = blockIdx.x;
    const int tid    = threadIdx.x;
    const int lane   = tid & 31;
    const int wave   = tid >> 5;   // 8 waves per block (wave32)

    const float* __restrict__ p = params + (size_t)sample * P_DIM;

    // stage input z[sample] (64 floats) into LDS
    if (tid < D_DIM) hA[tid] = z[sample * D_DIM + tid];
    __syncthreads();

    // L1: 64 -> 512, SiLU
    layer_wmma(p + OFF_W1, p + OFF_B1, hA, hB, D_DIM, H_DIM, true, wave, 8, lane);
    __syncthreads();
    // L2: 512 -> 512, SiLU
    layer_wmma(p + OFF_W2, p + OFF_B2, hB, hA, H_DIM, H_DIM, true, wave, 8, lane);
    __syncthreads();
    // L3: 512 -> 512, SiLU
    layer_wmma(p + OFF_W3, p + OFF_B3, hA, hB, H_DIM, H_DIM, true, wave, 8, lane);
    __syncthreads();
    // L4: 512 -> 64, linear, straight to global output
    layer_wmma(p + OFF_W4, p + OFF_B4, hB, out + (size_t)sample * D_DIM,
               H_DIM, D_DIM, false, wave, 8, lane);
}

extern "C" void kernel_launch(void* const* d_in, const int* in_sizes, int n_in,
                              void* d_out, int out_size, void* d_ws, size_t ws_size,
                              hipStream_t stream) {
    // inputs (setup_inputs order): t (unused by reference), z [B,64], params [B,P]
    const float* z      = (const float*)d_in[1];
    const float* params = (const float*)d_in[2];
    float* out          = (float*)d_out;   // [B,64] fp32

    (void)in_sizes; (void)n_in; (void)out_size; (void)d_ws; (void)ws_size;

    ode_persample_mlp_kernel<<<NB, 256, 0, stream>>>(z, params, out);
}